// KalmanFilter_14637248544856
// MI455X (gfx1250) — compile-verified
//
#include <hip/hip_runtime.h>
#include <hip/hip_bf16.h>

// Kalman filter: G=256 groups, T=512 steps, M=4 measures, S=32 state.
// One block per group, 128 threads = 4 waves; wave w owns 16x16 output tile
// (w>>1, w&1) of every 32x32 matmul. fp32 WMMA 16x16x4 for all S x S matmuls.
// Loop-invariant operands (F as A-frags, F^T and H as B-frags) are hoisted
// into registers before the serial time loop.

#define KF_G 256
#define KF_T 512
#define KF_M 4
#define KF_S 32

typedef __attribute__((ext_vector_type(2))) float v2f;
typedef __attribute__((ext_vector_type(8))) float v8f;

// D-tile of A(32x32)@B(32x32): A fragments in registers, B row-major in LDS.
__device__ __forceinline__ v8f kf_mm32_regA(const v2f* __restrict__ aF,
                                            const float* __restrict__ B,
                                            int tj, int lane) {
  v8f c = {};
  const int n  = lane & 15;
  const int kb = (lane >> 4) << 1;
#pragma unroll
  for (int kk = 0; kk < 8; ++kk) {
    const int k0 = kk * 4 + kb;
    v2f b;
    b.x = B[(k0)     * 32 + tj * 16 + n];
    b.y = B[(k0 + 1) * 32 + tj * 16 + n];
    c = __builtin_amdgcn_wmma_f32_16x16x4_f32(false, aF[kk], false, b,
                                              (short)0, c, false, false);
  }
  return c;
}

// D-tile of A(32x32)@B(32x32): A row-major in LDS, B fragments in registers.
__device__ __forceinline__ v8f kf_mm32_regB(const float* __restrict__ A,
                                            const v2f* __restrict__ bF,
                                            int ti, int lane) {
  v8f c = {};
  const int m  = lane & 15;
  const int kb = (lane >> 4) << 1;
#pragma unroll
  for (int kk = 0; kk < 8; ++kk) {
    const int k0 = kk * 4 + kb;
    v2f a;
    a.x = A[(ti * 16 + m) * 32 + k0];
    a.y = A[(ti * 16 + m) * 32 + k0 + 1];
    c = __builtin_amdgcn_wmma_f32_16x16x4_f32(false, a, false, bF[kk],
                                              (short)0, c, false, false);
  }
  return c;
}

// D-tile of A(32x32)@B(32x32): both operands row-major in LDS.
__device__ __forceinline__ v8f kf_mm32_tile(const float* __restrict__ A,
                                            const float* __restrict__ B,
                                            int ti, int tj, int lane) {
  v8f c = {};
  const int m  = lane & 15;
  const int kb = (lane >> 4) << 1;
#pragma unroll
  for (int kk = 0; kk < 8; ++kk) {
    const int k0 = kk * 4 + kb;
    v2f a, b;
    a.x = A[(ti * 16 + m) * 32 + k0];
    a.y = A[(ti * 16 + m) * 32 + k0 + 1];
    b.x = B[(k0)     * 32 + tj * 16 + m];
    b.y = B[(k0 + 1) * 32 + tj * 16 + m];
    c = __builtin_amdgcn_wmma_f32_16x16x4_f32(false, a, false, b,
                                              (short)0, c, false, false);
  }
  return c;
}

// Store C/D tile (row-striped: VGPR r -> rows r / r+8) to row-major LDS.
__device__ __forceinline__ void kf_store_tile(float* __restrict__ D,
                                              int ti, int tj, int lane, v8f c) {
  const int n  = lane & 15;
  const int hi = lane >> 4;
#pragma unroll
  for (int r = 0; r < 8; ++r) {
    const int row = ti * 16 + r + 8 * hi;
    D[row * 32 + tj * 16 + n] = c[r];
  }
}

__global__ __launch_bounds__(128)
void kf_kernel(const float* __restrict__ obs,   // [G,T,M]
               const float* __restrict__ F,     // [G,S,S]
               const float* __restrict__ H,     // [G,M,S]
               float* __restrict__ out) {       // means [G,T,S] ++ covs [G,T,S,S]
  __shared__ float F_s[KF_S * KF_S];
  __shared__ float Ft_s[KF_S * KF_S];
  __shared__ float H_s[KF_M * KF_S];
  __shared__ float cov_s[KF_S * KF_S];
  __shared__ float tmp_s[KF_S * KF_S];
  __shared__ float covp_s[KF_S * KF_S];
  __shared__ float akh_s[KF_S * KF_S];
  __shared__ float K_s[KF_S * KF_M];
  __shared__ float HP_s[KF_M * KF_S];
  __shared__ float Smat_s[KF_M * KF_M];
  __shared__ float mean_s[KF_S];
  __shared__ float meanp_s[KF_S];
  __shared__ float resid_s[KF_M];

  const int g    = blockIdx.x;
  const int tid  = threadIdx.x;
  const int lane = tid & 31;
  const int w    = tid >> 5;
  const int ti   = w >> 1;
  const int tj   = w & 1;

  const float* Fg   = F + (size_t)g * KF_S * KF_S;
  const float* Hg   = H + (size_t)g * KF_M * KF_S;
  const float* obsg = obs + (size_t)g * KF_T * KF_M;
  float* means_out  = out + (size_t)g * KF_T * KF_S;
  float* covs_out   = out + (size_t)KF_G * KF_T * KF_S
                          + (size_t)g * KF_T * KF_S * KF_S;

  // Load F (and transpose), H; init cov = I, mean = 0; emit t=0 outputs.
  for (int i = tid; i < KF_S * KF_S; i += 128) {
    const float v = Fg[i];
    const int r = i >> 5, c2 = i & 31;
    F_s[i] = v;
    Ft_s[c2 * 32 + r] = v;
    const float idv = (r == c2) ? 1.0f : 0.0f;
    cov_s[i] = idv;
    covs_out[i] = idv;
  }
  if (tid < KF_M * KF_S) H_s[tid] = Hg[tid];
  if (tid < KF_S) { mean_s[tid] = 0.0f; means_out[tid] = 0.0f; }
  __syncthreads();

  // Hoist loop-invariant WMMA operand fragments into registers.
  const int frag_mn = lane & 15;
  const int frag_kb = (lane >> 4) << 1;
  v2f fA[8];   // F as A-operand, tile row ti
  v2f ftB[8];  // F^T as B-operand, tile col tj
  v2f hB;      // H (4x32) as B-operand for K@H, tile col tj
#pragma unroll
  for (int kk = 0; kk < 8; ++kk) {
    const int k0 = kk * 4 + frag_kb;
    fA[kk].x  = F_s[(ti * 16 + frag_mn) * 32 + k0];
    fA[kk].y  = F_s[(ti * 16 + frag_mn) * 32 + k0 + 1];
    ftB[kk].x = Ft_s[(k0)     * 32 + tj * 16 + frag_mn];
    ftB[kk].y = Ft_s[(k0 + 1) * 32 + tj * 16 + frag_mn];
  }
  hB.x = H_s[(frag_kb)     * 32 + tj * 16 + frag_mn];
  hB.y = H_s[(frag_kb + 1) * 32 + tj * 16 + frag_mn];

  for (int t = 1; t < KF_T; ++t) {
    // ---- predict: tmp = F @ cov  (A in regs) -------------------------
    v8f c = kf_mm32_regA(fA, cov_s, tj, lane);
    kf_store_tile(tmp_s, ti, tj, lane, c);
    __syncthreads();

    // ---- cov_p = tmp @ F^T + Q (Q = I)  (B in regs) ------------------
    c = kf_mm32_regB(tmp_s, ftB, ti, lane);
    {
      const int n  = lane & 15;
      const int hi = lane >> 4;
#pragma unroll
      for (int r = 0; r < 8; ++r) {
        const int row = ti * 16 + r + 8 * hi;
        const int col = tj * 16 + n;
        c[r] += (row == col) ? 1.0f : 0.0f;
      }
    }
    kf_store_tile(covp_s, ti, tj, lane, c);
    __syncthreads();

    // ---- mean_p = F @ mean (wave0) ; HP = H @ cov_p (all 128) --------
    if (tid < KF_S) {
      float acc = 0.0f;
#pragma unroll 8
      for (int k = 0; k < KF_S; ++k) acc += F_s[tid * 32 + k] * mean_s[k];
      meanp_s[tid] = acc;
    }
    {
      const int m2 = tid >> 5, s2 = tid & 31;
      float acc = 0.0f;
#pragma unroll 8
      for (int k = 0; k < KF_S; ++k) acc += H_s[m2 * 32 + k] * covp_s[k * 32 + s2];
      HP_s[m2 * 32 + s2] = acc;
    }
    __syncthreads();

    // ---- S = HP @ H^T + R (R = I) ; resid = obs - H @ mean_p ---------
    if (tid < 16) {
      const int m2 = tid >> 2, n2 = tid & 3;
      float acc = (m2 == n2) ? 1.0f : 0.0f;
#pragma unroll 8
      for (int k = 0; k < KF_S; ++k) acc += HP_s[m2 * 32 + k] * H_s[n2 * 32 + k];
      Smat_s[tid] = acc;
    } else if (tid < 20) {
      const int m2 = tid - 16;
      float acc = obsg[(size_t)(t - 1) * KF_M + m2];
#pragma unroll 8
      for (int k = 0; k < KF_S; ++k) acc -= H_s[m2 * 32 + k] * meanp_s[k];
      resid_s[m2] = acc;
    }
    __syncthreads();

    // ---- solve S x = HP[:,lane]  (Gauss-Jordan, S is SPD, 4x4) -------
    // K[lane][m] = x[m];  mean_n = mean_p + K @ resid
    if (tid < KF_S) {
      float a[4][4];
#pragma unroll
      for (int i = 0; i < 4; ++i)
#pragma unroll
        for (int j = 0; j < 4; ++j) a[i][j] = Smat_s[i * 4 + j];
      float x[4];
#pragma unroll
      for (int m2 = 0; m2 < 4; ++m2) x[m2] = HP_s[m2 * 32 + tid];
#pragma unroll
      for (int p = 0; p < 4; ++p) {
        const float inv = 1.0f / a[p][p];
        x[p] *= inv;
#pragma unroll
        for (int c2 = 0; c2 < 4; ++c2) a[p][c2] *= inv;
#pragma unroll
        for (int r2 = 0; r2 < 4; ++r2) {
          if (r2 == p) continue;
          const float f2 = a[r2][p];
          x[r2] -= f2 * x[p];
#pragma unroll
          for (int c2 = 0; c2 < 4; ++c2) a[r2][c2] -= f2 * a[p][c2];
        }
      }
      float mn = meanp_s[tid];
#pragma unroll
      for (int m2 = 0; m2 < 4; ++m2) {
        K_s[tid * 4 + m2] = x[m2];
        mn += x[m2] * resid_s[m2];
      }
      mean_s[tid] = mn;
      means_out[(size_t)t * KF_S + tid] = mn;
    }
    __syncthreads();

    // ---- A = I - K @ H  (one 16x16x4 WMMA per wave, H frag in regs) --
    {
      v2f a;
      a.x = K_s[(ti * 16 + frag_mn) * 4 + frag_kb];
      a.y = K_s[(ti * 16 + frag_mn) * 4 + frag_kb + 1];
      v8f kh = {};
      kh = __builtin_amdgcn_wmma_f32_16x16x4_f32(false, a, false, hB,
                                                 (short)0, kh, false, false);
      const int n  = lane & 15;
      const int hi = lane >> 4;
#pragma unroll
      for (int r = 0; r < 8; ++r) {
        const int row = ti * 16 + r + 8 * hi;
        const int col = tj * 16 + n;
        akh_s[row * 32 + col] = ((row == col) ? 1.0f : 0.0f) - kh[r];
      }
    }
    __syncthreads();

    // ---- cov_n = A @ cov_p -------------------------------------------
    c = kf_mm32_tile(akh_s, covp_s, ti, tj, lane);
    kf_store_tile(cov_s, ti, tj, lane, c);
    __syncthreads();

    // Coalesced, vectorized global store of cov_n (2 x float4 per thread);
    // prefetch next observation row.
    {
      float* cdst = covs_out + (size_t)t * KF_S * KF_S;
      const float4* csrc4 = (const float4*)cov_s;
      float4* cdst4 = (float4*)cdst;
#pragma unroll
      for (int q = 0; q < 2; ++q) cdst4[tid * 2 + q] = csrc4[tid * 2 + q];
    }
    __builtin_prefetch(obsg + (size_t)t * KF_M, 0, 0);
    // cov_s is re-written only after further barriers next iteration,
    // so no extra barrier is needed here.
  }
}

extern "C" void kernel_launch(void* const* d_in, const int* in_sizes, int n_in,
                              void* d_out, int out_size, void* d_ws, size_t ws_size,
                              hipStream_t stream) {
  (void)in_sizes; (void)n_in; (void)out_size; (void)d_ws; (void)ws_size;
  const float* obs = (const float*)d_in[0];
  const float* F   = (const float*)d_in[1];
  const float* H   = (const float*)d_in[2];
  float* out       = (float*)d_out;
  kf_kernel<<<dim3(KF_G), dim3(128), 0, stream>>>(obs, F, H, out);
}